// KAN_Convolutional_Layer_78520592105536
// MI455X (gfx1250) — compile-verified
//
#include <hip/hip_runtime.h>
#include <hip/hip_bf16.h>

typedef __attribute__((ext_vector_type(2))) float v2f;
typedef __attribute__((ext_vector_type(8))) float v8f;

// ---- geometry -------------------------------------------------------------
// x: (16,16,64,64) f32, pad 1, K=3 -> out (16,128,64,64) f32
// K-dim of the fused GEMM: k = i*9 + t, i in [0,9) kernel pos, t in [0,9):
//   t<8 -> spline basis t  (weight = spline_weight[n,i,t]*spline_scaler[n,i])
//   t=8 -> silu(x)         (weight = base_weight[n,i])
// K padded 81 -> 84 (21 WMMA steps of K=4), M padded 8 convs -> 16.

#define STRIP_ROWS 16            // output rows per block
#define FROWS      18            // padded rows needed (16 + 2)
#define FPIX       (FROWS * 66)  // 1188 feature pixels per strip
#define KPAD       84

// LDS feature offset (in floats) for K index k, relative to pixel (h, w0+lane)
struct DofsT { int v[KPAD]; };
static constexpr DofsT make_dofs() {
    DofsT d{};
    for (int k = 0; k < KPAD; ++k) {
        if (k < 81) {
            int i = k / 9, t = k % 9;
            int di = i / 3, dj = i % 3;
            d.v[k] = (di * 66 + dj) * 9 + t;
        } else {
            d.v[k] = 0; // pad K: weight is zero, just read something in-bounds
        }
    }
    return d;
}
static constexpr DofsT DOFS = make_dofs();

// 8 cubic B-spline bases (efficient-kan grid: 12 knots, -2.2..2.2 step 0.4) + silu
__device__ __forceinline__ void kan_features(float x, float* f) {
    const float g0 = -2.2f, h = 0.4f;
    float b[11];
#pragma unroll
    for (int j = 0; j < 11; ++j) {
        float gj = g0 + h * (float)j;
        b[j] = (x >= gj && x < gj + h) ? 1.f : 0.f;
    }
#pragma unroll
    for (int k = 1; k <= 3; ++k) {
        float inv = 1.f / (h * (float)k);
#pragma unroll
        for (int j = 0; j + k < 11; ++j) {          // in-place safe left->right
            float gj  = g0 + h * (float)j;
            float gkr = gj + h * (float)(k + 1);    // grid[j+k+1]
            b[j] = (x - gj) * inv * b[j] + (gkr - x) * inv * b[j + 1];
        }
    }
#pragma unroll
    for (int t = 0; t < 8; ++t) f[t] = b[t];
    f[8] = x / (1.f + __expf(-x));                  // silu
}

__global__ __launch_bounds__(256) void kan_conv_wmma(
    const float* __restrict__ x,              // (16,16,64,64)
    const float* __restrict__ base_weight,    // (8,9)
    const float* __restrict__ spline_weight,  // (8,9,8)
    const float* __restrict__ spline_scaler,  // (8,9)
    float* __restrict__ out)                  // (16,128,64,64)
{
    __shared__ float FEAT[FPIX * 9];          // 42768 B
    __shared__ float Wl[16 * KPAD];           //  5376 B

    const int tid   = threadIdx.x;
    const int lane  = tid & 31;
    const int wave  = tid >> 5;
    const int bc    = blockIdx.x >> 2;        // fused (b,c) plane, 0..255
    const int strip = blockIdx.x & 3;         // 16-row strip, 0..3

    // ---- stage combined weight matrix W[16][84] ---------------------------
    for (int idx = tid; idx < 16 * KPAD; idx += 256) {
        int n = idx / KPAD, k = idx % KPAD;
        float v = 0.f;
        if (n < 8 && k < 81) {
            int i = k / 9, t = k % 9;
            v = (t < 8) ? spline_weight[(n * 9 + i) * 8 + t] * spline_scaler[n * 9 + i]
                        : base_weight[n * 9 + i];
        }
        Wl[idx] = v;
    }

    // ---- per-pixel features for this strip (computed once, reused 9x) ----
    for (int p = tid; p < FPIX; p += 256) {
        int r = p / 66, col = p % 66;
        int R = strip * STRIP_ROWS + r - 1;   // input row (padded coords -1)
        int Cc = col - 1;
        float xv = 0.f;
        if (R >= 0 && R < 64 && Cc >= 0 && Cc < 64)
            xv = x[((size_t)bc * 64 + R) * 64 + Cc];
        float f[9];
        kan_features(xv, f);
#pragma unroll
        for (int t = 0; t < 9; ++t) FEAT[p * 9 + t] = f[t];
    }
    __syncthreads();

    // ---- A fragments (weights) held in registers for the whole kernel ----
    // A 16x4 f32 layout: lane&15 = M row, VGPR v -> K = v + 2*(lane>=16)
    const int row  = lane & 15;
    const int hi   = lane >> 4;               // 0 or 1
    const int koff = hi * 2;
    v2f a[21];
#pragma unroll
    for (int s = 0; s < 21; ++s) {
        a[s].x = Wl[row * KPAD + 4 * s + koff];
        a[s].y = Wl[row * KPAD + 4 * s + koff + 1];
    }

    // ---- output tiles: 16 rows x 4 tiles = 64 tiles, 8 waves ------------
    for (int t = wave; t < STRIP_ROWS * 4; t += 8) {
        int h  = t >> 2;
        int w0 = (t & 3) << 4;
        int basep = (h * 66 + w0 + (lane & 15)) * 9;  // B col = lane&15

        // Phase 1: gather ALL B fragments for this tile into registers so the
        // WMMA chain below runs back-to-back behind one DS-count wait.
        v2f bfr[21];
#pragma unroll
        for (int s = 0; s < 21; ++s) {
            // B 4x16 f32 layout: lane&15 = N col, VGPR v -> K = v + 2*hi
            int d0 = hi ? DOFS.v[4 * s + 2] : DOFS.v[4 * s + 0];
            int d1 = hi ? DOFS.v[4 * s + 3] : DOFS.v[4 * s + 1];
            bfr[s].x = FEAT[basep + d0];
            bfr[s].y = FEAT[basep + d1];
        }

        // Phase 2: pure accumulating WMMA chain (D->C is the pipelined path).
        v8f acc = {0.f, 0.f, 0.f, 0.f, 0.f, 0.f, 0.f, 0.f};
#pragma unroll
        for (int s = 0; s < 21; ++s) {
            acc = __builtin_amdgcn_wmma_f32_16x16x4_f32(
                false, a[s], false, bfr[s], (short)0, acc, false, false);
        }

        // D layout: lanes 0..15 -> N = lane, VGPR n -> M = conv n (8 used)
        if (lane < 16) {
            int H0 = strip * STRIP_ROWS + h;
            size_t ob = ((size_t)bc * 8) * 4096 + (size_t)H0 * 64 + w0 + lane;
#pragma unroll
            for (int n = 0; n < 8; ++n) out[ob + (size_t)n * 4096] = acc[n];
        }
    }
}

extern "C" void kernel_launch(void* const* d_in, const int* in_sizes, int n_in,
                              void* d_out, int out_size, void* d_ws, size_t ws_size,
                              hipStream_t stream) {
    const float* x  = (const float*)d_in[0];
    const float* bw = (const float*)d_in[1];
    const float* sw = (const float*)d_in[2];
    const float* ss = (const float*)d_in[3];
    float* out = (float*)d_out;
    // 256 (b,c) planes x 4 row-strips
    kan_conv_wmma<<<dim3(256 * 4), dim3(256), 0, stream>>>(x, bw, ss != nullptr ? sw : sw, ss, out);
}